// TGCNCell_67989332295850
// MI455X (gfx1250) — compile-verified
//
#include <hip/hip_runtime.h>
#include <math.h>

typedef __attribute__((ext_vector_type(2))) float v2f;
typedef __attribute__((ext_vector_type(8))) float v8f;

#define KP 104   // K=101 zero-padded to a multiple of 4 (and 16B row alignment)

// ---------------- helpers ----------------

__device__ __forceinline__ void atomic_max_f32(float* addr, float v) {
  if (v >= 0.0f) atomicMax((int*)addr, __float_as_int(v));
  else           atomicMin((unsigned int*)addr, __float_as_uint(v));
}

__global__ void fill_f32(float* __restrict__ p, float v, int n) {
  int i = blockIdx.x * blockDim.x + threadIdx.x;
  if (i < n) p[i] = v;
}

// x[bn][c] (row stride KP): c==0 -> inputs; 1..U -> (gate?)*state; U+1..KP-1 -> 0
__global__ void build_x(const float* __restrict__ inp, const float* __restrict__ st,
                        const float* __restrict__ gate, int CG,
                        float* __restrict__ x, int tot, int U_) {
  int i = blockIdx.x * blockDim.x + threadIdx.x;
  if (i >= tot) return;
  int c  = i % KP;
  int bn = i / KP;
  float v = 0.0f;
  if (c == 0) v = inp[bn];
  else if (c <= U_) {
    v = st[bn * U_ + (c - 1)];
    if (gate) v *= gate[bn * CG + (c - 1)];
  }
  x[i] = v;
}

// Zero-K-padded weight repack: Wp[KP x Nout] col block [colOff, colOff+Nc) <- W[K x Nc]
__global__ void pad_w(const float* __restrict__ W, const float* __restrict__ bsrc,
                      float* __restrict__ Wp, float* __restrict__ bp,
                      int K, int Nc, int Nout, int colOff) {
  int i = blockIdx.x * blockDim.x + threadIdx.x;
  int tot = KP * Nc;
  if (i < tot) {
    int k = i / Nc, j = i % Nc;
    Wp[k * Nout + colOff + j] = (k < K) ? W[k * Nc + j] : 0.0f;
  }
  if (i < Nc) bp[colOff + i] = bsrc[i];
}

// ---------------- WMMA GEMM: Out[M,Nout] = X[M,KP(pad)] @ Wp[KP,Nout] + bp ----------------
// grid.x = N-tile, grid.y = group of 8 M-tiles; B tile staged in LDS, shared by all 8 waves.
__global__ __launch_bounds__(256)
void gemm_bias_wmma(const float* __restrict__ X, const float* __restrict__ Wp,
                    const float* __restrict__ bp, float* __restrict__ Out,
                    int Mtiles, int Nout) {
  __shared__ float Bs[KP * 16];
  const int colBase = blockIdx.x << 4;

  // cooperative B-tile load (clamped col: OOB D columns are never stored)
  for (int i = threadIdx.x; i < KP * 16; i += blockDim.x) {
    int k = i >> 4, j = i & 15;
    int col = colBase + j;
    if (col >= Nout) col = Nout - 1;
    Bs[i] = Wp[k * Nout + col];
  }
  __syncthreads();

  int tm = blockIdx.y * 8 + (threadIdx.x >> 5);
  if (tm >= Mtiles) return;              // wave-uniform -> EXEC stays all-ones
  int lane = threadIdx.x & 31;
  int kh   = lane >> 4;                  // selects K pair {0,1} vs {2,3}
  int l16  = lane & 15;
  const float* __restrict__ xrow = X + (size_t)((tm << 4) + l16) * KP;

  v8f acc = {};
#pragma unroll
  for (int k0 = 0; k0 < KP; k0 += 4) {
    int ka = k0 + (kh << 1);
    v2f a = *(const v2f*)(xrow + ka);     // aligned b64 load (KP*4 = 416B row stride)
    v2f b;
    b.x = Bs[ka * 16 + l16];              // B VGPR0: row 2h   (conflict-free)
    b.y = Bs[(ka + 1) * 16 + l16];        // B VGPR1: row 2h+1
    acc = __builtin_amdgcn_wmma_f32_16x16x4_f32(
        /*neg_a=*/false, a, /*neg_b=*/false, b,
        /*c_mod=*/(short)0, acc, /*reuse_a=*/false, /*reuse_b=*/false);
  }

  int col = colBase + l16;
  if (col < Nout) {
    float bv = bp[col];
    int rbase = (tm << 4) + (kh << 3);    // C/D: VGPR r -> row r (+8 for high half)
#pragma unroll
    for (int r = 0; r < 8; ++r)
      Out[(size_t)(rbase + r) * Nout + col] = acc[r] + bv;
  }
}

// ---------------- edge phase (one wave32 per edge) ----------------
// xlr rows hold [xl (C) | xr (C)] with row stride CS = 2*C.

__global__ void edge_logits(const float* __restrict__ xlr, const float* __restrict__ att,
                            const int* __restrict__ src, const int* __restrict__ dst,
                            float* __restrict__ eout, float* __restrict__ smax,
                            int CS, int C, int Ef, int N_, int E_, int totWaves) {
  int w = blockIdx.x * (blockDim.x >> 5) + (threadIdx.x >> 5);
  if (w >= totWaves) return;
  int lane = threadIdx.x & 31;
  int b = w / Ef, e = w % Ef;
  int s = (e < E_) ? src[e] : (e - E_);
  int d = (e < E_) ? dst[e] : (e - E_);
  const float* pl = xlr + ((size_t)b * N_ + s) * CS;       // xl[src]
  const float* pr = xlr + ((size_t)b * N_ + d) * CS + C;   // xr[dst]
  float acc = 0.0f;
  for (int c = lane; c < C; c += 32) {
    float m = pl[c] + pr[c];
    m = (m > 0.0f) ? m : 0.2f * m;                         // leaky_relu 0.2
    acc += m * att[c];
  }
#pragma unroll
  for (int off = 16; off > 0; off >>= 1) acc += __shfl_xor(acc, off, 32);
  if (lane == 0) {
    eout[(size_t)b * Ef + e] = acc;
    atomic_max_f32(&smax[b * N_ + d], acc);
  }
}

__global__ void edge_exp(float* __restrict__ e, const float* __restrict__ smax,
                         float* __restrict__ den, const int* __restrict__ dst,
                         int Ef, int N_, int E_, int tot) {
  int i = blockIdx.x * blockDim.x + threadIdx.x;
  if (i >= tot) return;
  int b = i / Ef, ed = i % Ef;
  int d = (ed < E_) ? dst[ed] : (ed - E_);
  int nd = b * N_ + d;
  float v = expf(e[i] - smax[nd]);
  e[i] = v;
  atomicAdd(&den[nd], v);
}

__global__ void edge_scatter(const float* __restrict__ ea, const float* __restrict__ den,
                             const float* __restrict__ xlr,
                             const int* __restrict__ src, const int* __restrict__ dst,
                             float* __restrict__ out,
                             int CS, int C, int Ef, int N_, int E_, int totWaves) {
  int w = blockIdx.x * (blockDim.x >> 5) + (threadIdx.x >> 5);
  if (w >= totWaves) return;
  int lane = threadIdx.x & 31;
  int b = w / Ef, e = w % Ef;
  int s = (e < E_) ? src[e] : (e - E_);
  int d = (e < E_) ? dst[e] : (e - E_);
  float alpha = ea[(size_t)b * Ef + e] / den[b * N_ + d];
  const float* pl = xlr + ((size_t)b * N_ + s) * CS;       // xl[src]
  float* po = out + ((size_t)b * N_ + d) * C;
  for (int c = lane; c < C; c += 32)
    atomicAdd(&po[c], alpha * pl[c]);
}

// ---------------- pointwise epilogues ----------------

__global__ void add_bias2_sigmoid(float* __restrict__ g, const float* __restrict__ b1,
                                  const float* __restrict__ b2, int tot, int C) {
  int i = blockIdx.x * blockDim.x + threadIdx.x;
  if (i >= tot) return;
  int c = i % C;
  float v = g[i] + b1[c] + b2[c];
  g[i] = 1.0f / (1.0f + expf(-v));
}

__global__ void final_state(const float* __restrict__ out2, const float* __restrict__ val,
                            const float* __restrict__ st, const float* __restrict__ bg2,
                            const float* __restrict__ bias1, float* __restrict__ outp,
                            int tot, int U_, int C1) {
  int i = blockIdx.x * blockDim.x + threadIdx.x;
  if (i >= tot) return;
  int j  = i % U_;
  int bn = i / U_;
  float u = val[(size_t)bn * C1 + U_ + j];                 // update gate
  float c = tanhf(out2[i] + bg2[j] + bias1[j]);
  outp[i] = u * st[i] + (1.0f - u) * c;
}

// ---------------- driver ----------------

extern "C" void kernel_launch(void* const* d_in, const int* in_sizes, int n_in,
                              void* d_out, int out_size, void* d_ws, size_t ws_size,
                              hipStream_t stream) {
  (void)in_sizes; (void)n_in; (void)out_size; (void)ws_size;
  const int Bc = 2, Nn = 20000, Ee = 160000, Uu = 100, INc = 101;
  const int C1 = 200, C2 = 100;
  const int Ef = Ee + Nn;
  const int M  = Bc * Nn;                 // 40000 (16-divisible)
  const int Mtiles = M / 16;              // 2500

  const float* inputs = (const float*)d_in[0];
  const float* state  = (const float*)d_in[1];
  const int*   src    = (const int*)d_in[2];
  const int*   dst    = (const int*)d_in[3];
  const float* Wl1 = (const float*)d_in[4];
  const float* bl1 = (const float*)d_in[5];
  const float* Wr1 = (const float*)d_in[6];
  const float* br1 = (const float*)d_in[7];
  const float* att1 = (const float*)d_in[8];
  const float* bg1  = (const float*)d_in[9];
  const float* bias0 = (const float*)d_in[10];
  const float* Wl2 = (const float*)d_in[11];
  const float* bl2 = (const float*)d_in[12];
  const float* Wr2 = (const float*)d_in[13];
  const float* br2 = (const float*)d_in[14];
  const float* att2 = (const float*)d_in[15];
  const float* bg2  = (const float*)d_in[16];
  const float* bias1 = (const float*)d_in[17];

  float* ws = (float*)d_ws;
  size_t off = 0;
  float* x    = ws + off; off += (size_t)M * KP;     // padded features
  float* Wp1  = ws + off; off += (size_t)KP * 2*C1;  // [Wl1|Wr1] padded
  float* bp1  = ws + off; off += 2 * C1;
  float* Wp2  = ws + off; off += (size_t)KP * 2*C2;  // [Wl2|Wr2] padded
  float* bp2  = ws + off; off += 2 * C2;
  float* xlr  = ws + off; off += (size_t)M * 2*C1;   // [xl|xr] (layer2 reuses prefix)
  float* g    = ws + off; off += (size_t)M * C1;     // g1 -> sigmoid gates r,u
  float* o2   = ws + off; off += (size_t)M * C2;     // layer-2 aggregate
  float* ebuf = ws + off; off += (size_t)Bc * Ef;
  float* smax = ws + off; off += (size_t)M;
  float* den  = ws + off; off += (size_t)M;

  const int TB = 256;
  auto cdiv = [](int a, int b) { return (a + b - 1) / b; };
  const int nwav  = Bc * Ef;
  const int blksw = cdiv(nwav, TB / 32);
  const int totx  = M * KP;

  // ---- layer 1: GATv2(101 -> 200) ----
  pad_w<<<cdiv(KP * C1, TB), TB, 0, stream>>>(Wl1, bl1, Wp1, bp1, INc, C1, 2*C1, 0);
  pad_w<<<cdiv(KP * C1, TB), TB, 0, stream>>>(Wr1, br1, Wp1, bp1, INc, C1, 2*C1, C1);
  build_x<<<cdiv(totx, TB), TB, 0, stream>>>(inputs, state, nullptr, 0, x, totx, Uu);
  {
    dim3 gg(cdiv(2*C1, 16), cdiv(Mtiles, 8));       // 25 x 313
    gemm_bias_wmma<<<gg, TB, 0, stream>>>(x, Wp1, bp1, xlr, Mtiles, 2*C1);
  }
  fill_f32<<<cdiv(M, TB), TB, 0, stream>>>(smax, -INFINITY, M);
  fill_f32<<<cdiv(M, TB), TB, 0, stream>>>(den, 0.0f, M);
  fill_f32<<<cdiv(M * C1, TB), TB, 0, stream>>>(g, 0.0f, M * C1);
  edge_logits<<<blksw, TB, 0, stream>>>(xlr, att1, src, dst, ebuf, smax, 2*C1, C1, Ef, Nn, Ee, nwav);
  edge_exp<<<cdiv(nwav, TB), TB, 0, stream>>>(ebuf, smax, den, dst, Ef, Nn, Ee, nwav);
  edge_scatter<<<blksw, TB, 0, stream>>>(ebuf, den, xlr, src, dst, g, 2*C1, C1, Ef, Nn, Ee, nwav);
  add_bias2_sigmoid<<<cdiv(M * C1, TB), TB, 0, stream>>>(g, bg1, bias0, M * C1, C1);

  // ---- layer 2: GATv2(101 -> 100) on [inputs, r*state] ----
  pad_w<<<cdiv(KP * C2, TB), TB, 0, stream>>>(Wl2, bl2, Wp2, bp2, INc, C2, 2*C2, 0);
  pad_w<<<cdiv(KP * C2, TB), TB, 0, stream>>>(Wr2, br2, Wp2, bp2, INc, C2, 2*C2, C2);
  build_x<<<cdiv(totx, TB), TB, 0, stream>>>(inputs, state, g, C1, x, totx, Uu);
  {
    dim3 gg(cdiv(2*C2, 16), cdiv(Mtiles, 8));       // 13 x 313
    gemm_bias_wmma<<<gg, TB, 0, stream>>>(x, Wp2, bp2, xlr, Mtiles, 2*C2);
  }
  fill_f32<<<cdiv(M, TB), TB, 0, stream>>>(smax, -INFINITY, M);
  fill_f32<<<cdiv(M, TB), TB, 0, stream>>>(den, 0.0f, M);
  fill_f32<<<cdiv(M * C2, TB), TB, 0, stream>>>(o2, 0.0f, M * C2);
  edge_logits<<<blksw, TB, 0, stream>>>(xlr, att2, src, dst, ebuf, smax, 2*C2, C2, Ef, Nn, Ee, nwav);
  edge_exp<<<cdiv(nwav, TB), TB, 0, stream>>>(ebuf, smax, den, dst, Ef, Nn, Ee, nwav);
  edge_scatter<<<blksw, TB, 0, stream>>>(ebuf, den, xlr, src, dst, o2, 2*C2, C2, Ef, Nn, Ee, nwav);

  // ---- GRU-style update ----
  final_state<<<cdiv(M * Uu, TB), TB, 0, stream>>>(o2, g, state, bg2, bias1,
                                                   (float*)d_out, M * Uu, Uu, C1);
}